// PositionAwareAttention_44074954391730
// MI455X (gfx1250) — compile-verified
//
#include <hip/hip_runtime.h>
#include <hip/hip_bf16.h>

#define B_DIM  4
#define S_DIM  256
#define DIN    1024
#define DFEAT  1024
#define DATTN  256

typedef float v2f __attribute__((ext_vector_type(2)));
typedef float v8f __attribute__((ext_vector_type(8)));

// ---------------------------------------------------------------------------
// gfx1250 feature shims (all guarded so compile never regresses)
// ---------------------------------------------------------------------------
#if __has_builtin(__builtin_amdgcn_global_load_async_to_lds_b128)
#define HAVE_ASYNC 1
#else
#define HAVE_ASYNC 0
#endif

#if HAVE_ASYNC
// From the clang diagnostic, the builtin takes:
//   param0: int __attribute__((vector_size(16))) AS(1)*   (global src)
//   param1: same vector type in AS(3)                      (LDS dst)
typedef int asyncv4 __attribute__((vector_size(16)));
typedef __attribute__((address_space(1))) asyncv4* async_gp;
typedef __attribute__((address_space(3))) asyncv4* async_lp;
#endif

// 16B coalesced async copy global -> LDS (per-lane), tracked by ASYNCcnt.
__device__ __forceinline__ void copy16_to_lds(float* ldst, const float* gsrc) {
#if HAVE_ASYNC
  __builtin_amdgcn_global_load_async_to_lds_b128(
      (async_gp)(asyncv4*)(void*)const_cast<float*>(gsrc),
      (async_lp)(asyncv4*)(void*)ldst,
      0, 0);
#else
  *(float4*)ldst = *(const float4*)gsrc;
#endif
}

__device__ __forceinline__ void async_wait0() {
#if HAVE_ASYNC
#if __has_builtin(__builtin_amdgcn_s_wait_asynccnt)
  __builtin_amdgcn_s_wait_asynccnt(0);
#else
  asm volatile("s_wait_asynccnt 0" ::: "memory");
#endif
#endif
}

// gfx1250 has hardware v_tanh_f32; else 1-trans exp2-based tanh.
__device__ __forceinline__ float fast_tanh(float x) {
#if __has_builtin(__builtin_amdgcn_tanhf)
  return __builtin_amdgcn_tanhf(x);
#elif __has_builtin(__builtin_amdgcn_tanh_f32)
  return __builtin_amdgcn_tanh_f32(x);
#else
  x = fminf(fmaxf(x, -15.0f), 15.0f);
  float e = __builtin_amdgcn_exp2f(x * 2.8853900817779268f); // e^(2x)
  return (e - 1.0f) * __builtin_amdgcn_rcpf(e + 1.0f);
#endif
}

// ---------------------------------------------------------------------------
// K0: f_proj[b,a] = query[b,:] . Ww[a,:]   (tiny)
// ---------------------------------------------------------------------------
__global__ void paa_fproj(const float* __restrict__ query,
                          const float* __restrict__ Ww,
                          float* __restrict__ F) {
  const int b = blockIdx.x;
  const int a = threadIdx.x;
  const float4* q4 = (const float4*)(query + b * DFEAT);
  const float4* w4 = (const float4*)(Ww + (size_t)a * DFEAT);
  float acc = 0.f;
#pragma unroll 4
  for (int d = 0; d < DFEAT / 4; ++d) {
    float4 q = q4[d], w = w4[d];
    acc = fmaf(q.x, w.x, acc);
    acc = fmaf(q.y, w.y, acc);
    acc = fmaf(q.z, w.z, acc);
    acc = fmaf(q.w, w.w, acc);
  }
  F[b * DATTN + a] = acc;
}

// ---------------------------------------------------------------------------
// K1: WMMA f32 16x16x4 projections, LDS-staged via async copies.
// Block = 4 waves: shared 16-row hidden tile (m0..m0+15) x K-chunk 64,
// shared 64-row weight tile (nblk0..nblk0+63) x K-chunk 64.
// Row stride padded to 68 floats => ds_load_b64 fragment reads are
// bank-conflict free (bank = (4*lane + k) % 64).
//   z==0: U = hidden.Wu^T + bu + F ;  z==1: V = hidden.Wv^T
// ---------------------------------------------------------------------------
#define KC   64
#define LDA  68   // padded row stride (floats)

__global__ void paa_proj(const float* __restrict__ hidden,
                         const float* __restrict__ Wu,
                         const float* __restrict__ bu,
                         const float* __restrict__ Wv,
                         const float* __restrict__ F,
                         float* __restrict__ U,
                         float* __restrict__ V) {
  __shared__ float lA[16 * LDA];   // hidden tile  [m][k]
  __shared__ float lW[64 * LDA];   // weight tile  [n][k]

  const int lane = threadIdx.x;
  const int ty   = threadIdx.y;
  const int tid  = ty * 32 + lane;         // 0..127
  const int nl   = lane & 15;
  const int hi   = lane >> 4;              // K sub-offset 2*hi
  const int m0    = blockIdx.x * 16;       // rows over B*S
  const int nblk0 = blockIdx.y * 64;       // block's 64 output cols
  const bool isU  = (blockIdx.z == 0);
  const float* Wsel = isU ? Wu : Wv;

  v8f c = {};
  for (int k0 = 0; k0 < DIN; k0 += KC) {
    // ---- stage A tile: 16 rows x 64 floats = 256 x 16B segments ----
#pragma unroll
    for (int r = 0; r < 2; ++r) {
      const int idx = tid + 128 * r;       // 0..255
      const int row = idx >> 4, seg = idx & 15;
      copy16_to_lds(lA + row * LDA + seg * 4,
                    hidden + (size_t)(m0 + row) * DIN + k0 + seg * 4);
    }
    // ---- stage W tile: 64 rows x 64 floats = 1024 x 16B segments ----
#pragma unroll
    for (int r = 0; r < 8; ++r) {
      const int idx = tid + 128 * r;       // 0..1023
      const int row = idx >> 4, seg = idx & 15;
      copy16_to_lds(lW + row * LDA + seg * 4,
                    Wsel + (size_t)(nblk0 + row) * DIN + k0 + seg * 4);
    }
    async_wait0();
    __syncthreads();

    // ---- 16 WMMA steps over this K-chunk ----
    const float* arow = lA + nl * LDA + 2 * hi;
    const float* brow = lW + (ty * 16 + nl) * LDA + 2 * hi;
#pragma unroll
    for (int kk = 0; kk < KC; kk += 4) {
      v2f a = *(const v2f*)(arow + kk);    // A[m][k]  (ds_load_b64)
      v2f b = *(const v2f*)(brow + kk);    // B[k][n] = W[n][k]
      c = __builtin_amdgcn_wmma_f32_16x16x4_f32(false, a, false, b,
                                                (short)0, c, false, false);
    }
    __syncthreads();
  }

  const int n0 = nblk0 + ty * 16;
  float bias = 0.f;
  if (isU) {
    const int b_ = m0 / S_DIM;             // tile never crosses batch
    bias = bu[n0 + nl] + F[b_ * DATTN + n0 + nl];
  }
  float* dst = isU ? U : V;
#pragma unroll
  for (int r = 0; r < 8; ++r) {
    const int row = m0 + r + 8 * hi;       // C/D: lanes16-31 hold M=r+8
    dst[(size_t)row * DATTN + n0 + nl] = c[r] + bias;
  }
}

// ---------------------------------------------------------------------------
// K2: the trans-bound bottleneck. One block per (b,i); thread j:
//   score(j) = wt . tanh(U[b,j,:] + V[b,i,:]) + bt, then block softmax.
// ---------------------------------------------------------------------------
__global__ void paa_score_softmax(const float* __restrict__ U,
                                  const float* __restrict__ V,
                                  const float* __restrict__ wt,
                                  const float* __restrict__ btp,
                                  const int* __restrict__ mask,
                                  float* __restrict__ AW) {
  __shared__ float vsh[DATTN];
  __shared__ float wsh[DATTN];
  __shared__ float red[8];

  const int j = threadIdx.x;
  const int i = blockIdx.x;
  const int b = blockIdx.y;

  vsh[j] = V[((size_t)(b * S_DIM + i)) * DATTN + j];
  wsh[j] = wt[j];
  __syncthreads();

  float score;
  if (mask[b * S_DIM + j] < 1) {
    score = -1e30f;
  } else {
    const float4* u4 = (const float4*)(U + ((size_t)(b * S_DIM + j)) * DATTN);
    float acc = 0.f;
#pragma unroll 4
    for (int a4 = 0; a4 < DATTN / 4; ++a4) {
      const float4 u = u4[a4];
      const int a = a4 * 4;
      acc = fmaf(wsh[a + 0], fast_tanh(u.x + vsh[a + 0]), acc);
      acc = fmaf(wsh[a + 1], fast_tanh(u.y + vsh[a + 1]), acc);
      acc = fmaf(wsh[a + 2], fast_tanh(u.z + vsh[a + 2]), acc);
      acc = fmaf(wsh[a + 3], fast_tanh(u.w + vsh[a + 3]), acc);
    }
    score = acc + btp[0];
  }

  float m = score;
#pragma unroll
  for (int off = 16; off >= 1; off >>= 1) m = fmaxf(m, __shfl_xor(m, off, 32));
  const int w = j >> 5;
  if ((j & 31) == 0) red[w] = m;
  __syncthreads();
  m = red[0];
#pragma unroll
  for (int t = 1; t < 8; ++t) m = fmaxf(m, red[t]);

  const float e = __builtin_amdgcn_exp2f((score - m) * 1.44269504088896340736f);
  __syncthreads();

  float s = e;
#pragma unroll
  for (int off = 16; off >= 1; off >>= 1) s += __shfl_xor(s, off, 32);
  if ((j & 31) == 0) red[w] = s;
  __syncthreads();
  s = red[0];
#pragma unroll
  for (int t = 1; t < 8; ++t) s += red[t];

  AW[((size_t)(b * S_DIM + i)) * S_DIM + j] = e / s;
}

// ---------------------------------------------------------------------------
// K3: out[b,i,d] = sum_j AW[b,i,j] * hidden[b,j,d].
// All 4 waves of a block share (b, d-tile): stage hidden B-tile
// [64 j x 16 d] in LDS per K-chunk via async copies; A rows read straight
// from AW (row-major float2, coalesced). lB[k][n] reads are conflict-free.
// ---------------------------------------------------------------------------
__global__ void paa_outgemm(const float* __restrict__ AW,
                            const float* __restrict__ hidden,
                            float* __restrict__ out) {
  __shared__ float lB[KC * 16];            // [k][n], 4KB

  const int lane = threadIdx.x;
  const int ty   = threadIdx.y;
  const int tid  = ty * 32 + lane;         // 0..127
  const int nl   = lane & 15;
  const int hi   = lane >> 4;
  const int n0 = blockIdx.x * 16;                      // d tile (shared)
  const int m0 = (blockIdx.y * 4 + ty) * 16;           // i tile (per wave)
  const int b = blockIdx.z;

  const float* hb   = hidden + (size_t)b * S_DIM * DIN;
  const float* Arow = AW + ((size_t)(b * S_DIM + m0 + nl)) * S_DIM + 2 * hi;

  v8f c = {};
  for (int k0 = 0; k0 < S_DIM; k0 += KC) {
    // stage 64 j-rows x 16 d floats = 256 x 16B segments
#pragma unroll
    for (int r = 0; r < 2; ++r) {
      const int idx = tid + 128 * r;       // 0..255
      const int row = idx >> 2, seg = idx & 3;
      copy16_to_lds(lB + row * 16 + seg * 4,
                    hb + (size_t)(k0 + row) * DIN + n0 + seg * 4);
    }
    async_wait0();
    __syncthreads();

#pragma unroll
    for (int kk = 0; kk < KC; kk += 4) {
      v2f a = *(const v2f*)(Arow + k0 + kk);           // A[i][j]
      v2f bb;
      bb.x = lB[(kk + 2 * hi) * 16 + nl];              // B[j][d]
      bb.y = lB[(kk + 2 * hi + 1) * 16 + nl];
      c = __builtin_amdgcn_wmma_f32_16x16x4_f32(false, a, false, bb,
                                                (short)0, c, false, false);
    }
    __syncthreads();
  }

  float* orow = out + (size_t)b * S_DIM * DIN;
#pragma unroll
  for (int r = 0; r < 8; ++r) {
    orow[(size_t)(m0 + r + 8 * hi) * DIN + n0 + nl] = c[r];
  }
}

// ---------------------------------------------------------------------------
extern "C" void kernel_launch(void* const* d_in, const int* in_sizes, int n_in,
                              void* d_out, int out_size, void* d_ws, size_t ws_size,
                              hipStream_t stream) {
  const float* hidden = (const float*)d_in[0];  // [4,256,1024]
  const int*   mask   = (const int*)  d_in[1];  // [4,256]
  const float* query  = (const float*)d_in[2];  // [4,1024]
  const float* Wu     = (const float*)d_in[3];  // [256,1024]
  const float* bu     = (const float*)d_in[4];  // [256]
  const float* Wv     = (const float*)d_in[5];  // [256,1024]
  const float* Ww     = (const float*)d_in[6];  // [256,1024]
  const float* wt     = (const float*)d_in[7];  // [256]
  const float* bt     = (const float*)d_in[8];  // scalar
  float* out = (float*)d_out;

  float* F  = (float*)d_ws;
  float* U  = F + B_DIM * DATTN;
  float* V  = U + (size_t)B_DIM * S_DIM * DATTN;
  float* AW = V + (size_t)B_DIM * S_DIM * DATTN;

  paa_fproj<<<dim3(B_DIM), dim3(DATTN), 0, stream>>>(query, Ww, F);

  // 64 row-tiles x (4 blocks * 4 waves) col-tiles x {U,V}
  paa_proj<<<dim3((B_DIM * S_DIM) / 16, 4, 2), dim3(32, 4), 0, stream>>>(
      hidden, Wu, bu, Wv, F, U, V);

  paa_score_softmax<<<dim3(S_DIM, B_DIM), dim3(S_DIM), 0, stream>>>(
      U, V, wt, bt, mask, AW);

  paa_outgemm<<<dim3(DIN / 16, 4, B_DIM), dim3(32, 4), 0, stream>>>(
      AW, hidden, out);
}